// KeepHighResolutionModulePartSeg_75136157876259
// MI455X (gfx1250) — compile-verified
//
#include <hip/hip_runtime.h>

// MI455X / gfx1250 implementation.
// Key algebraic optimization: scatter-average and Linear commute, so we project
// features to 64 channels FIRST with fp32 WMMA GEMMs (V_WMMA_F32_16X16X4_F32),
// then scatter-average in 64-dim space. This turns a 256MB intermediate into
// an 8MB one; the whole pipeline becomes HBM-bandwidth bound (~100MB total).
// All bulk elementwise traffic uses 128-bit per-lane accesses; BN stats are
// reduced through LDS (ds_add_f32) before hitting global atomics.
//
// Workspace layout (floats), total ~10.7M floats (~41 MB):
//   gbuf  [2097152]  projected features (max B*1024*64)
//   kidx  [131072]   knn indices (int, max B*512*8)
//   sums  [4194304]  scatter accumulator / GEMM output [B*2048, 64]
//   cnt   [65536]    scatter counts [B*2048]
//   stats [128]      per-channel sum & sumsq
//   acc   [4194304]  running f0 + sum of branch outputs

typedef __attribute__((ext_vector_type(2))) float v2f;
typedef __attribute__((ext_vector_type(8))) float v8f;

#define BATCH   32
#define NPT     2048
#define CH      64
#define KNN_K   8
#define EPS_BN  1e-5f
#define SLOPE   0.2f
#define NROWS   (BATCH * NPT)          /* 65536   */
#define NELEM   (NROWS * CH)           /* 4194304 */

// ---------------- vectorized elementwise helpers (b128 path) ----------------
__global__ void copy4_kernel(const float4* __restrict__ src, float4* __restrict__ dst, int n4) {
  int i = blockIdx.x * blockDim.x + threadIdx.x;
  if (i < n4) dst[i] = src[i];
}

__global__ void zero4_kernel(float4* __restrict__ p, int n4) {
  int i = blockIdx.x * blockDim.x + threadIdx.x;
  if (i < n4) { float4 z = {0.f, 0.f, 0.f, 0.f}; p[i] = z; }
}

// ---------------- kNN: 8 nearest of 2048 points per query ----------------
__global__ void knn_kernel(const float* __restrict__ xyz_all,  // [B, 2048, 3]
                           const float* __restrict__ xyz_q,    // [B, S, 3]
                           int S, int* __restrict__ out) {     // [B, S, 8]
  int t = blockIdx.x * blockDim.x + threadIdx.x;
  if (t >= BATCH * S) return;
  int b = t / S;
  const float* q = xyz_q + (size_t)t * 3;
  float qx = q[0], qy = q[1], qz = q[2];
  const float* P = xyz_all + (size_t)b * NPT * 3;
  float bd[KNN_K]; int bi[KNN_K];
#pragma unroll
  for (int j = 0; j < KNN_K; ++j) { bd[j] = 3.0e38f; bi[j] = 0; }
  for (int n = 0; n < NPT; ++n) {
    float dx = qx - P[n*3+0], dy = qy - P[n*3+1], dz = qz - P[n*3+2];
    float d = dx*dx + dy*dy + dz*dz;
    if (d < bd[KNN_K-1]) {
      bd[KNN_K-1] = d; bi[KNN_K-1] = n;
#pragma unroll
      for (int j = KNN_K-1; j > 0; --j) {   // one bubble pass: new elem sinks in
        if (bd[j] < bd[j-1]) {
          float td = bd[j]; bd[j] = bd[j-1]; bd[j-1] = td;
          int   ti = bi[j]; bi[j] = bi[j-1]; bi[j-1] = ti;
        }
      }
    }
  }
#pragma unroll
  for (int j = 0; j < KNN_K; ++j) out[(size_t)t * KNN_K + j] = bi[j];
}

// ---------------- fp32 WMMA GEMM: D[M,64] = A[M,K] @ W[64,K]^T ----------------
// One wave = one 16x16 output tile; 4 waves/block cover the 64 output columns.
// A-frag (16x4 f32): lanes 0-15 -> M rows, VGPR pair -> K={kb,kb+1}, kb=2*(lane>>4).
// B-frag (4x16): lane&15 -> N col, same K mapping; B[k][n] = W[n][k].
// D: 8 VGPRs, row M = v + 8*(lane>>4), col N = lane&15.
__global__ void gemm64_wmma(const float* __restrict__ A,   // [M, K]
                            const float* __restrict__ W,   // [64, K]
                            float* __restrict__ D,         // [M, 64]
                            int K) {
  const int row0 = blockIdx.x * 16;
  const int lane = threadIdx.x & 31;
  const int wave = threadIdx.x >> 5;            // col tile 0..3
  const int m    = lane & 15;
  const int kb   = (lane >> 4) << 1;
  const float* arow = A + (size_t)(row0 + m) * K;
  const float* brow = W + (size_t)(wave * 16 + m) * K;   // n == lane&15 == m
  v8f acc = {};
#pragma unroll 4
  for (int kk = 0; kk < K; kk += 4) {
    v2f a, b;
    a.x = arow[kk + kb]; a.y = arow[kk + kb + 1];
    b.x = brow[kk + kb]; b.y = brow[kk + kb + 1];
    acc = __builtin_amdgcn_wmma_f32_16x16x4_f32(false, a, false, b,
                                                (short)0, acc, false, false);
  }
  const int hi  = lane >> 4;
  const int col = wave * 16 + (lane & 15);
#pragma unroll
  for (int v = 0; v < 8; ++v)
    D[(size_t)(row0 + v + 8*hi) * CH + col] = acc[v];
}

// ---------------- scatter-add of projected features ----------------
// thread t -> (e = (b*S+s)*8+k, cg = 16-channel group 0..3)
__global__ void scatter_kernel(const float* __restrict__ g,   // [B*S, 64]
                               const int*  __restrict__ idx,  // [B*S, 8]
                               float* __restrict__ sums,      // [B*2048, 64]
                               float* __restrict__ cnt,       // [B*2048]
                               int S) {
  int t = blockIdx.x * blockDim.x + threadIdx.x;
  int total = BATCH * S * KNN_K * 4;
  if (t >= total) return;
  int cg  = t & 3;
  int e   = t >> 2;
  int src = e >> 3;            // b*S + s
  int b   = src / S;
  int n   = idx[e];
  const float* gp = g + (size_t)src * CH + cg * 16;
  float* sp = sums + ((size_t)b * NPT + n) * CH + cg * 16;
  if (cg == 0) atomicAdd(&cnt[b * NPT + n], 1.0f);
#pragma unroll
  for (int c = 0; c < 16; ++c) atomicAdd(&sp[c], gp[c]);
}

// ---------------- divide by count, add bias, gather BN stats ----------------
// block = 64 threads = (16 channel-groups of 4ch x 4 row-groups); block covers
// 256 rows with float4 (b128) accesses. Partial sums reduced in LDS
// (ds_add_f32), then 128 global atomics per block.
__global__ void finalize_stats(float* __restrict__ y,          // [65536, 64] in-place
                               const float* __restrict__ cnt,  // [65536] or nullptr
                               const float* __restrict__ bias, // [64]
                               float* __restrict__ stats) {    // [128] sum,sumsq
  __shared__ float red[2 * CH];
  const int tid = threadIdx.x;          // 0..63
  const int cg  = (tid & 15) * 4;       // channel base (multiple of 4)
  const int rg  = tid >> 4;             // 0..3
  const int r0  = blockIdx.x * 256;
  const float4 bv = *reinterpret_cast<const float4*>(bias + cg);
  float sx = 0.f, sy = 0.f, sz = 0.f, sw = 0.f;
  float qx = 0.f, qy = 0.f, qz = 0.f, qw = 0.f;
  for (int i = rg; i < 256; i += 4) {
    const int row = r0 + i;
    float4* yp = reinterpret_cast<float4*>(y + (size_t)row * CH + cg);
    float4 v = *yp;
    if (cnt) {
      float n = cnt[row];
      float sc = (n > 0.f) ? (1.0f / n) : 1.0f;
      v.x *= sc; v.y *= sc; v.z *= sc; v.w *= sc;
    }
    v.x += bv.x; v.y += bv.y; v.z += bv.z; v.w += bv.w;
    *yp = v;
    sx += v.x; sy += v.y; sz += v.z; sw += v.w;
    qx += v.x*v.x; qy += v.y*v.y; qz += v.z*v.z; qw += v.w*v.w;
  }
  red[tid] = 0.f; red[tid + 64] = 0.f;
  __syncthreads();
  atomicAdd(&red[cg + 0], sx); atomicAdd(&red[cg + 1], sy);
  atomicAdd(&red[cg + 2], sz); atomicAdd(&red[cg + 3], sw);
  atomicAdd(&red[CH + cg + 0], qx); atomicAdd(&red[CH + cg + 1], qy);
  atomicAdd(&red[CH + cg + 2], qz); atomicAdd(&red[CH + cg + 3], qw);
  __syncthreads();
  atomicAdd(&stats[tid], red[tid]);
  atomicAdd(&stats[tid + 64], red[tid + 64]);
}

// ---------------- BatchNorm + LeakyReLU, accumulate or residual-out ----------------
// float4 per thread: 4 consecutive channels of one row (all b128 traffic).
__global__ void bn_act_kernel(const float4* __restrict__ y,
                              const float* __restrict__ stats,
                              const float* __restrict__ gamma,
                              const float* __restrict__ beta,
                              float4* __restrict__ dst,
                              const float4* __restrict__ residual) { // null => dst += act
  int i = blockIdx.x * blockDim.x + threadIdx.x;
  if (i >= NELEM / 4) return;
  const int c = (i * 4) & (CH - 1);       // multiple of 4
  const float inv = 1.0f / (float)NROWS;
  const float4 s1 = *reinterpret_cast<const float4*>(stats + c);
  const float4 s2 = *reinterpret_cast<const float4*>(stats + CH + c);
  const float4 gm = *reinterpret_cast<const float4*>(gamma + c);
  const float4 bt = *reinterpret_cast<const float4*>(beta + c);
  float4 v = y[i];
  float m, va, t;
  m = s1.x * inv; va = s2.x * inv - m*m;
  t = (v.x - m) * rsqrtf(va + EPS_BN) * gm.x + bt.x; v.x = (t >= 0.f) ? t : SLOPE * t;
  m = s1.y * inv; va = s2.y * inv - m*m;
  t = (v.y - m) * rsqrtf(va + EPS_BN) * gm.y + bt.y; v.y = (t >= 0.f) ? t : SLOPE * t;
  m = s1.z * inv; va = s2.z * inv - m*m;
  t = (v.z - m) * rsqrtf(va + EPS_BN) * gm.z + bt.z; v.z = (t >= 0.f) ? t : SLOPE * t;
  m = s1.w * inv; va = s2.w * inv - m*m;
  t = (v.w - m) * rsqrtf(va + EPS_BN) * gm.w + bt.w; v.w = (t >= 0.f) ? t : SLOPE * t;
  if (residual) {
    float4 r = residual[i];
    v.x += r.x; v.y += r.y; v.z += r.z; v.w += r.w;
    dst[i] = v;
  } else {
    float4 d = dst[i];
    d.x += v.x; d.y += v.y; d.z += v.z; d.w += v.w;
    dst[i] = d;
  }
}

extern "C" void kernel_launch(void* const* d_in, const int* in_sizes, int n_in,
                              void* d_out, int out_size, void* d_ws, size_t ws_size,
                              hipStream_t stream) {
  const float* f0   = (const float*)d_in[1];
  const float* f1   = (const float*)d_in[2];
  const float* f2   = (const float*)d_in[3];
  const float* f3   = (const float*)d_in[4];
  const float* f4   = (const float*)d_in[5];
  const int*   knn1 = (const int*)d_in[6];
  const float* xyz0 = (const float*)d_in[7];
  const float* xyz2 = (const float*)d_in[8];
  const float* xyz3 = (const float*)d_in[9];
  const float* xyz4 = (const float*)d_in[10];
  const float *Wt[5], *bs[5], *gm[5], *bt[5];
  for (int j = 0; j < 5; ++j) {
    Wt[j] = (const float*)d_in[11 + 4*j + 0];
    bs[j] = (const float*)d_in[11 + 4*j + 1];
    gm[j] = (const float*)d_in[11 + 4*j + 2];
    bt[j] = (const float*)d_in[11 + 4*j + 3];
  }
  float* out = (float*)d_out;

  float* ws    = (float*)d_ws;
  float* gbuf  = ws;                                   // 2097152 floats
  int*   kidx  = (int*)(ws + 2097152);                 // 131072 ints
  float* sums  = ws + 2097152 + 131072;                // NELEM
  float* cnt   = sums + (size_t)NELEM;                 // NROWS
  float* stats = cnt + NROWS;                          // 128
  float* acc   = stats + 2*CH;                         // NELEM

  dim3 b256(256);

  // acc = f0  (b128 copy)
  copy4_kernel<<<(NELEM/4 + 255)/256, b256, 0, stream>>>(
      (const float4*)f0, (float4*)acc, NELEM/4);

  const float* feats[4] = {f1, f2, f3, f4};
  const float* qxyz[4]  = {nullptr, xyz2, xyz3, xyz4};
  const int    Ss[4]    = {1024, 512, 256, 128};
  const int    Ks[4]    = {128, 256, 512, 1024};

  for (int st = 0; st < 4; ++st) {
    const int S = Ss[st], K = Ks[st];
    const int M = BATCH * S;
    const int* idx = knn1;
    if (st > 0) {
      knn_kernel<<<(BATCH*S + 255)/256, b256, 0, stream>>>(xyz0, qxyz[st], S, kidx);
      idx = kidx;
    }
    // project first (Linear commutes with averaging): g = f @ W^T
    gemm64_wmma<<<M/16, dim3(128), 0, stream>>>(feats[st], Wt[st], gbuf, K);
    // zero sums + cnt + stats (contiguous, count divisible by 4)
    zero4_kernel<<<((NELEM + NROWS + 2*CH)/4 + 255)/256, b256, 0, stream>>>(
        (float4*)sums, (NELEM + NROWS + 2*CH)/4);
    scatter_kernel<<<(BATCH*S*KNN_K*4 + 255)/256, b256, 0, stream>>>(
        gbuf, idx, sums, cnt, S);
    finalize_stats<<<NROWS/256, dim3(64), 0, stream>>>(sums, cnt, bs[st], stats);
    bn_act_kernel<<<(NELEM/4 + 255)/256, b256, 0, stream>>>(
        (const float4*)sums, stats, gm[st], bt[st], (float4*)acc, nullptr);
  }

  // final: z = acc @ W0^T + b0 -> BN -> LeakyReLU -> + f0
  gemm64_wmma<<<NROWS/16, dim3(128), 0, stream>>>(acc, Wt[4], sums, CH);
  zero4_kernel<<<1, dim3(32), 0, stream>>>((float4*)stats, (2*CH)/4);
  finalize_stats<<<NROWS/256, dim3(64), 0, stream>>>(sums, nullptr, bs[4], stats);
  bn_act_kernel<<<(NELEM/4 + 255)/256, b256, 0, stream>>>(
      (const float4*)sums, stats, gm[4], bt[4], (float4*)out, (const float4*)f0);
}